// MyAugmentationPipeline_27118423507032
// MI455X (gfx1250) — compile-verified
//
#include <hip/hip_runtime.h>
#include <hip/hip_bf16.h>
#include <stdint.h>

// Problem geometry (fixed by setup_inputs):
//   input: (1, 16, 64, 160, 160) f32, mask: (64,160,160) f32, flags: int32[5]
#define Wd   160
#define Hd   160
#define Dd   64
#define Cc   16
#define SPAT (Wd * Hd * Dd)          // 1,638,400 voxels per channel

// Native 128-bit vector (HIP's float4 is a class type and is rejected by
// __builtin_nontemporal_store).
typedef float __attribute__((ext_vector_type(4))) f32x4;

// ---------------------------------------------------------------------------
// gfx1250 async global->LDS path (ASYNCcnt-tracked). Guarded so the file
// always compiles; fallback is a plain register gather.
// ---------------------------------------------------------------------------
#if defined(__HIP_DEVICE_COMPILE__) && __has_builtin(__builtin_amdgcn_global_load_async_to_lds_b32)
#define HAVE_ASYNC_LDS 1
#else
#define HAVE_ASYNC_LDS 0
#endif

#if HAVE_ASYNC_LDS
typedef __attribute__((address_space(1))) int as1_int;   // printed as "__device__ int"
typedef __attribute__((address_space(3))) int as3_int;   // printed as "__shared__ int"

__device__ __forceinline__ void g2l_b32(const float* g, float* l) {
  // (global src, LDS dst, imm offset, imm cpol)
  __builtin_amdgcn_global_load_async_to_lds_b32((as1_int*)g, (as3_int*)l, 0, 0);
}
__device__ __forceinline__ void wait_async0() {
#if __has_builtin(__builtin_amdgcn_s_wait_asynccnt)
  __builtin_amdgcn_s_wait_asynccnt(0);
#else
  asm volatile("s_wait_asynccnt 0" ::: "memory");
#endif
}
#endif

// ---------------------------------------------------------------------------
// Composed gather map.
// The three warp rotations are exact signed permutations with half-integer
// centers, so trilinear interpolation degenerates to a single tap with
// zero-fill out of range. Gather maps (applied in REVERSE of forward order:
// Rs, Re, Rn, flipH, flipW):
//   Rs: (sx,sy,sz) <- (111 - sz, sx,       111 - sy)
//   Re: (sx,sy,sz) <- (111 - sz, 159 - sy, 111 - sx)
//   Rn: (sx,sy,sz) <- (111 - sz, 159 - sx, sy - 48 )
// Each warp stage zero-fills when its source coordinate leaves the volume.
// ---------------------------------------------------------------------------
__device__ __forceinline__ bool in_vol(int sx, int sy, int sz) {
  return (unsigned)sx < (unsigned)Wd &&
         (unsigned)sy < (unsigned)Hd &&
         (unsigned)sz < (unsigned)Dd;
}

__global__ __launch_bounds__(Wd) void aug_pipeline_kernel(
    const float* __restrict__ in,     // 16 * SPAT
    const float* __restrict__ mask,   // SPAT
    const int*   __restrict__ flags,  // 5
    float*       __restrict__ out_inp,   // 16 * SPAT
    float*       __restrict__ out_mask)  // SPAT
{
  const int row = blockIdx.x;          // 0 .. Dd*Hd-1
  const int z   = row / Hd;
  const int y   = row - z * Hd;
  const int x   = threadIdx.x;         // 0 .. 159

  const int f0 = flags[0], f1 = flags[1], f2 = flags[2],
            f3 = flags[3], f4 = flags[4];

  int  sx = x, sy = y, sz = z;
  bool valid = true;

  if (f4 > 0) {                        // Rs (applied last forward -> first in gather)
    int nx = 111 - sz, ny = sx, nz = 111 - sy;
    sx = nx; sy = ny; sz = nz;
    valid = valid && in_vol(sx, sy, sz);
  }
  if (f3 > 0) {                        // Re
    int nx = 111 - sz, ny = 159 - sy, nz = 111 - sx;
    sx = nx; sy = ny; sz = nz;
    valid = valid && in_vol(sx, sy, sz);
  }
  if (f2 > 0) {                        // Rn
    int nx = 111 - sz, ny = 159 - sx, nz = sy - 48;
    sx = nx; sy = ny; sz = nz;
    valid = valid && in_vol(sx, sy, sz);
  }
  if (f1 > 0) sy = (Hd - 1) - sy;      // flip H
  if (f0 > 0) sx = (Wd - 1) - sx;      // flip W

  // Clamp to a safe address when invalid; the value is zeroed afterwards.
  const int srcoff = valid ? ((sz * Hd + sy) * Wd + sx) : 0;
  const int dstrow = row * Wd;

#if HAVE_ASYNC_LDS
  // ---- async gather into LDS, then b128 drain with NT stores --------------
  __shared__ __align__(16) float tile[(Cc + 1) * Wd];   // 10.9 KB of 320 KB/WGP

  #pragma unroll
  for (int c = 0; c < Cc; ++c)
    g2l_b32(in + c * SPAT + srcoff, &tile[c * Wd + x]);
  g2l_b32(mask + srcoff, &tile[Cc * Wd + x]);

  wait_async0();        // our wave's async copies have landed in LDS
  __syncthreads();      // all waves' copies visible

  if (!valid) {         // zero-fill out-of-range gather lanes
    #pragma unroll
    for (int c = 0; c <= Cc; ++c) tile[c * Wd + x] = 0.0f;
  }
  __syncthreads();

  // Drain: 17 rows * 40 float4 = 680 b128 transfers, NT so the output stream
  // does not evict the (L2-resident) input volume.
  const int nQuads = (Cc + 1) * (Wd / 4);               // 680
  for (int i = x; i < nQuads; i += Wd) {
    const int c = i / (Wd / 4);
    const int g = i - c * (Wd / 4);
    f32x4 v = reinterpret_cast<const f32x4*>(tile)[i];
    float* base = (c < Cc) ? (out_inp + c * SPAT + dstrow)
                           : (out_mask + dstrow);
    __builtin_nontemporal_store(v, reinterpret_cast<f32x4*>(base) + g);
  }
#else
  // ---- fallback: register gather + NT scalar stores -----------------------
  #pragma unroll
  for (int c = 0; c < Cc; ++c) {
    float v = in[c * SPAT + srcoff];
    v = valid ? v : 0.0f;
    __builtin_nontemporal_store(v, out_inp + c * SPAT + dstrow + x);
  }
  float m = mask[srcoff];
  m = valid ? m : 0.0f;
  __builtin_nontemporal_store(m, out_mask + dstrow + x);
#endif
}

extern "C" void kernel_launch(void* const* d_in, const int* in_sizes, int n_in,
                              void* d_out, int out_size, void* d_ws, size_t ws_size,
                              hipStream_t stream) {
  const float* in    = (const float*)d_in[0];
  const float* mask  = (const float*)d_in[1];
  const int*   flags = (const int*)d_in[2];

  float* out_inp  = (float*)d_out;
  float* out_mask = (float*)d_out + (size_t)Cc * SPAT;   // outputs concatenated

  dim3 grid(Dd * Hd);   // 10240 rows
  dim3 block(Wd);       // 160 threads = 5 wave32
  hipLaunchKernelGGL(aug_pipeline_kernel, grid, block, 0, stream,
                     in, mask, flags, out_inp, out_mask);
}